// FFTConv2d_8942121910411
// MI455X (gfx1250) — compile-verified
//
#include <hip/hip_runtime.h>

typedef __attribute__((ext_vector_type(2))) float v2f;
typedef __attribute__((ext_vector_type(8))) float v8f;
typedef __attribute__((ext_vector_type(4))) unsigned int v4u;
typedef __attribute__((ext_vector_type(4))) int v4i;
typedef __attribute__((ext_vector_type(8))) int v8i;

#define NB      16
#define CIN     32
#define COUT    32
#define HH      256
#define WW      256
#define KH      31
#define KW      31
#define HOUT    226
#define WOUT    226
#define TILE_X  128
#define SROW    (TILE_X + 32)   // 160 floats: 128 outputs + 31 halo + 1 pad
#define NITER   (CIN * KH)      // 992 staged (c,u) slices

// Issue one TDM 2D tile load: global -> LDS. 6-arg clang-23 builtin form.
// D# group0: count=1, lds_addr, 57-bit global addr, type=2.
// D# group1: data_size=4B, tensor_dim0/1, tile_dim0/1 (tile_dim2=0), dim0 stride.
__device__ __forceinline__ void tdm_load_2d(unsigned lds_off,
                                            unsigned long long ga,
                                            unsigned td0, unsigned td1,
                                            unsigned tile0, unsigned tile1,
                                            unsigned long long stride0)
{
    v4u g0;
    g0.x = 1u;                                               // count=1 (valid)
    g0.y = lds_off;                                          // LDS byte address
    g0.z = (unsigned)(ga & 0xFFFFFFFFull);                   // global_addr[31:0]
    g0.w = (unsigned)((ga >> 32) & 0x1FFFFFFull) | (2u << 30); // addr[56:32] | type=2

    v8i g1;
    g1[0] = (int)(2u << 16);                                 // data_size = 4 bytes
    g1[1] = (int)((td0 & 0xFFFFu) << 16);                    // tensor_dim0[15:0]
    g1[2] = (int)(((td1 & 0xFFFFu) << 16) | ((td0 >> 16) & 0xFFFFu));
    g1[3] = (int)(((tile0 & 0xFFFFu) << 16) | ((td1 >> 16) & 0xFFFFu));
    g1[4] = (int)(tile1 & 0xFFFFu);                          // tile_dim1 (tile_dim2=0)
    g1[5] = (int)(stride0 & 0xFFFFFFFFull);                  // dim0_stride[31:0]
    g1[6] = (int)((stride0 >> 32) & 0xFFFFull);              // dim0_stride[47:32]
    g1[7] = 0;

    v4i gz4 = {0, 0, 0, 0};
    v8i gz8 = {0, 0, 0, 0, 0, 0, 0, 0};
    __builtin_amdgcn_tensor_load_to_lds(g0, g1, gz4, gz4, gz8, 0);
}

__global__ __launch_bounds__(256)
void fftconv_igemm_f32_wmma_tdm(const float* __restrict__ sig,
                                const float* __restrict__ wgt,
                                const float* __restrict__ bias,
                                float* __restrict__ out)
{
    __shared__ float sW[2][COUT * KW];   // double-buffered weight slice (packed stride 31)
    __shared__ float sS[2][SROW];        // double-buffered signal row segment

    const int tid  = threadIdx.x;
    const int lane = tid & 31;
    const int wave = tid >> 5;
    const int wm   = wave >> 2;    // 0..1 : cout half
    const int wn   = wave & 3;     // 0..3 : 32-wide x strip (two 16x16 N tiles)
    const int half = lane >> 4;
    const int l16  = lane & 15;

    const int x0 = blockIdx.x * TILE_X;
    const int y  = blockIdx.y;
    const int b  = blockIdx.z;

    // LDS byte offsets (generic-pointer low 32 bits == LDS byte address)
    const unsigned sW_off0 = (unsigned)(unsigned long long)&sW[0][0];
    const unsigned sW_off1 = (unsigned)(unsigned long long)&sW[1][0];
    const unsigned sS_off0 = (unsigned)(unsigned long long)&sS[0][0];
    const unsigned sS_off1 = (unsigned)(unsigned long long)&sS[1][0];

    // iteration it = c*KH + u ; weight slice base advances KW floats per it
    unsigned long long wgt_ga = (unsigned long long)wgt;
    unsigned long long sig_ga =
        (unsigned long long)(sig + (((size_t)(b * CIN) * HH + y) * WW + x0));
    const unsigned sig_td0 = (unsigned)(WW - x0);  // TDM OOB returns zero -> edge clamp

    // prologue: stage slice 0 into buffer 0 (one TDM per descriptor, one wave each)
    if (wave == 0)
        tdm_load_2d(sW_off0, wgt_ga, KW, COUT, KW, COUT,
                    (unsigned long long)(CIN * KH * KW));
    if (wave == 1)
        tdm_load_2d(sS_off0, sig_ga, sig_td0, 1, SROW, 1, (unsigned long long)WW);

    v8f acc0 = {};
    v8f acc1 = {};

    const int cobase = (wm * 16 + l16) * KW;  // A row (M) base in packed sW
    const int nb0    = wn * 32 + l16;         // B column (N) base for first subtile

    int u = 0;
    for (int it = 0; it < NITER; ++it) {
        const int p = it & 1;
        if (wave < 2) __builtin_amdgcn_s_wait_tensorcnt(0);
        __syncthreads();  // buf p ready for all; buf 1-p free to refill

        if (it + 1 < NITER) {
            wgt_ga += KW * 4;
            if (++u == KH) { u = 0; sig_ga += (unsigned long long)(HH - KH + 1) * WW * 4; }
            else           {        sig_ga += (unsigned long long)WW * 4;                 }
            if (wave == 0)
                tdm_load_2d(p ? sW_off0 : sW_off1, wgt_ga, KW, COUT, KW, COUT,
                            (unsigned long long)(CIN * KH * KW));
            if (wave == 1)
                tdm_load_2d(p ? sS_off0 : sS_off1, sig_ga, sig_td0, 1, SROW, 1,
                            (unsigned long long)WW);
        }

        const float* sWp = &sW[p][0];
        const float* sSp = &sS[p][0];

        // v = 0..27 in 7 full K=4 chunks; A reused across both N subtiles
        #pragma unroll
        for (int kc = 0; kc < 7; ++kc) {
            const int kb = kc * 4 + half * 2;
            v2f a, b0, b1;
            a.x  = sWp[cobase + kb];
            a.y  = sWp[cobase + kb + 1];
            b0.x = sSp[nb0 + kb];
            b0.y = sSp[nb0 + kb + 1];
            b1.x = sSp[nb0 + 16 + kb];
            b1.y = sSp[nb0 + 16 + kb + 1];
            acc0 = __builtin_amdgcn_wmma_f32_16x16x4_f32(false, a, false, b0,
                                                         (short)0, acc0, false, false);
            acc1 = __builtin_amdgcn_wmma_f32_16x16x4_f32(false, a, false, b1,
                                                         (short)0, acc1, false, false);
        }
        {   // tail chunk: v = 28..30, v=31 annihilated via a.y = 0
            const int kb = 28 + half * 2;
            v2f a, b0, b1;
            a.x  = sWp[cobase + kb];
            a.y  = half ? 0.0f : sWp[cobase + 29];
            b0.x = sSp[nb0 + kb];
            b0.y = sSp[nb0 + kb + 1];
            b1.x = sSp[nb0 + 16 + kb];
            b1.y = sSp[nb0 + 16 + kb + 1];
            acc0 = __builtin_amdgcn_wmma_f32_16x16x4_f32(false, a, false, b0,
                                                         (short)0, acc0, false, false);
            acc1 = __builtin_amdgcn_wmma_f32_16x16x4_f32(false, a, false, b1,
                                                         (short)0, acc1, false, false);
        }
    }

    // epilogue: D 16x16 layout (VGPR r -> M = half*8 + r), add bias, mask x edge
    const int co = wm * 16 + half * 8;
    const int xA = x0 + wn * 32 + l16;
    if (xA < WOUT) {
        #pragma unroll
        for (int r = 0; r < 8; ++r)
            out[((b * COUT + co + r) * HOUT + y) * WOUT + xA] = acc0[r] + bias[co + r];
    }
    const int xB = xA + 16;
    if (xB < WOUT) {
        #pragma unroll
        for (int r = 0; r < 8; ++r)
            out[((b * COUT + co + r) * HOUT + y) * WOUT + xB] = acc1[r] + bias[co + r];
    }
}

extern "C" void kernel_launch(void* const* d_in, const int* in_sizes, int n_in,
                              void* d_out, int out_size, void* d_ws, size_t ws_size,
                              hipStream_t stream) {
    const float* sig  = (const float*)d_in[0];
    const float* wgt  = (const float*)d_in[1];
    const float* bias = (const float*)d_in[2];
    float* out = (float*)d_out;

    dim3 grid((WOUT + TILE_X - 1) / TILE_X, HOUT, NB);  // 2 x 226 x 16
    fftconv_igemm_f32_wmma_tdm<<<grid, 256, 0, stream>>>(sig, wgt, bias, out);
}